// Decompressor_67886253081135
// MI455X (gfx1250) — compile-verified
//
#include <hip/hip_runtime.h>
#include <cstddef>

// Problem constants (from the reference).
#define DD     2048      // D_IN: hidden/input width
#define SEQLEN 256       // sequence length
#define DOUT   1024      // D_OUT

typedef __attribute__((ext_vector_type(2))) float v2f;
typedef __attribute__((ext_vector_type(8))) float v8f;

// ---------------------------------------------------------------------------
// W_sum = W_ih + W_hh  (run once; enables halved per-step weight traffic,
// and 64 MB result stays resident in the 192 MB L2 across all 255 steps).
// ---------------------------------------------------------------------------
__global__ __launch_bounds__(256) void wsum_kernel(const float4* __restrict__ a,
                                                   const float4* __restrict__ b,
                                                   float4* __restrict__ o) {
  size_t i = (size_t)blockIdx.x * 256 + threadIdx.x;
  float4 x = a[i];
  float4 y = b[i];
  o[i] = make_float4(x.x + y.x, x.y + y.y, x.z + y.z, x.w + y.w);
}

// ---------------------------------------------------------------------------
// One LSTM step. gates[j + g*D] = dot(W1 row, v) [+ dot(W2 row, v)] + b.
// One wave32 per hidden unit j (its 4 gate rows i,f,g,o); 8 waves per block;
// 256 blocks cover all 2048 hidden units. v (8 KB) is staged in LDS and
// broadcast-read; weight rows are streamed with coalesced float4 loads
// (L2-resident after the first step). Shuffle butterfly reduces each dot.
// ---------------------------------------------------------------------------
__global__ __launch_bounds__(256) void lstm_step_kernel(
    const float* __restrict__ W1,    // [4D, D] applied to v
    const float* __restrict__ W2,    // [4D, D] applied to v (nullptr if unused)
    const float* __restrict__ v,     // [D]  (x at t=0, h_{t-1} otherwise)
    const float* __restrict__ b_ih,  // [4D]
    const float* __restrict__ b_hh,  // [4D]
    float* __restrict__ c,           // [D] in/out cell state
    float* __restrict__ h_out,       // [D] = seq row t (also next step's input)
    int use_w2, int first) {
  __shared__ float sv[DD];
  for (int i = threadIdx.x; i < DD; i += 256) sv[i] = v[i];
  __syncthreads();

  const int wave = threadIdx.x >> 5;
  const int lane = threadIdx.x & 31;
  const int j    = blockIdx.x * 8 + wave;          // hidden unit 0..2047

  const size_t r0 = (size_t)j * DD;                // row j start
  const size_t rs = (size_t)DD * DD;               // gate-block stride (2048 rows)

  float acc[4] = {0.f, 0.f, 0.f, 0.f};

  for (int k0 = lane * 4; k0 < DD; k0 += 128) {
    float4 xv = *(const float4*)(sv + k0);
#pragma unroll
    for (int g = 0; g < 4; ++g) {
      const float* wp = W1 + r0 + (size_t)g * rs + k0;
      __builtin_prefetch(wp + 128, 0, 0);          // global_prefetch_b8 next chunk
      float4 w = *(const float4*)wp;
      acc[g] = fmaf(w.x, xv.x, acc[g]);
      acc[g] = fmaf(w.y, xv.y, acc[g]);
      acc[g] = fmaf(w.z, xv.z, acc[g]);
      acc[g] = fmaf(w.w, xv.w, acc[g]);
    }
    if (use_w2) {
#pragma unroll
      for (int g = 0; g < 4; ++g) {
        float4 w = *(const float4*)(W2 + r0 + (size_t)g * rs + k0);
        acc[g] = fmaf(w.x, xv.x, acc[g]);
        acc[g] = fmaf(w.y, xv.y, acc[g]);
        acc[g] = fmaf(w.z, xv.z, acc[g]);
        acc[g] = fmaf(w.w, xv.w, acc[g]);
      }
    }
  }

  // wave32 butterfly reduction
#pragma unroll
  for (int off = 16; off > 0; off >>= 1) {
#pragma unroll
    for (int g = 0; g < 4; ++g) acc[g] += __shfl_down(acc[g], off, 32);
  }

  if (lane == 0) {
    float bi = b_ih[j]          + b_hh[j];
    float bf = b_ih[j + DD]     + b_hh[j + DD];
    float bg = b_ih[j + 2 * DD] + b_hh[j + 2 * DD];
    float bo = b_ih[j + 3 * DD] + b_hh[j + 3 * DD];
    float ig = 1.0f / (1.0f + expf(-(acc[0] + bi)));
    float fg = 1.0f / (1.0f + expf(-(acc[1] + bf)));
    float gg = tanhf(acc[2] + bg);
    float og = 1.0f / (1.0f + expf(-(acc[3] + bo)));
    float c_old = first ? 0.0f : c[j];
    float cn = fg * c_old + ig * gg;
    c[j] = cn;
    h_out[j] = og * tanhf(cn);
  }
}

// ---------------------------------------------------------------------------
// out[256,1024] = seq[256,2048] @ W_out.T + b_out using V_WMMA_F32_16X16X4_F32
// (full fp32 precision). One wave per 16x16 C tile; K-loop in steps of 4.
// Lane layouts per CDNA5 ISA §7.12.2:
//   A 16x4 : lanes 0-15 -> M=lane, K={0,1}; lanes 16-31 -> M=lane-16, K={2,3}
//   B 4x16 : lanes 0-15 -> N=lane, K={0,1}; lanes 16-31 -> N=lane-16, K={2,3}
//   C/D    : VGPR r, lanes 0-15 -> M=r, N=lane; lanes 16-31 -> M=r+8, N=lane-16
// ---------------------------------------------------------------------------
__global__ __launch_bounds__(256) void out_proj_kernel(
    const float* __restrict__ seq,   // [256, 2048]
    const float* __restrict__ Wout,  // [1024, 2048]; B[k][n] = Wout[n][k]
    const float* __restrict__ bout,  // [1024]
    float* __restrict__ out) {       // [256, 1024]
  const int K = DD;
  const int wave = threadIdx.x >> 5;
  const int lane = threadIdx.x & 31;
  const int tile = blockIdx.x * 8 + wave;  // 0..1023
  const int tn = tile & 63;                // 64 tiles over N=1024
  const int tm = tile >> 6;                // 16 tiles over M=256

  const int mn    = lane & 15;             // row (A) / col (B) within tile
  const int khalf = (lane >> 4) * 2;       // K sub-offset: 0 or 2

  const float* ap = seq  + (size_t)(tm * 16 + mn) * K + khalf;
  const float* bp = Wout + (size_t)(tn * 16 + mn) * K + khalf;

  v8f acc = {0.f, 0.f, 0.f, 0.f, 0.f, 0.f, 0.f, 0.f};
#pragma unroll 4
  for (int kk = 0; kk < K; kk += 4) {
    v2f a = *(const v2f*)(ap + kk);
    v2f b = *(const v2f*)(bp + kk);
    acc = __builtin_amdgcn_wmma_f32_16x16x4_f32(
        /*neg_a=*/false, a, /*neg_b=*/false, b,
        /*c_mod=*/(short)0, acc, /*reuse_a=*/false, /*reuse_b=*/false);
  }

  const int nOut  = tn * 16 + (lane & 15);
  const int mBase = tm * 16 + ((lane >> 4) ? 8 : 0);
  const float bb  = bout[nOut];
#pragma unroll
  for (int r = 0; r < 8; ++r) {
    out[(size_t)(mBase + r) * DOUT + nOut] = acc[r] + bb;
  }
}

// ---------------------------------------------------------------------------
extern "C" void kernel_launch(void* const* d_in, const int* in_sizes, int n_in,
                              void* d_out, int out_size, void* d_ws, size_t ws_size,
                              hipStream_t stream) {
  (void)in_sizes; (void)n_in; (void)out_size;

  const float* x     = (const float*)d_in[0];
  const float* W_ih  = (const float*)d_in[1];
  const float* W_hh  = (const float*)d_in[2];
  const float* b_ih  = (const float*)d_in[3];
  const float* b_hh  = (const float*)d_in[4];
  const float* W_out = (const float*)d_in[5];
  const float* b_out = (const float*)d_in[6];
  // d_in[7] = seq_len (device scalar); fixed at 256 per the reference config.

  // Workspace layout: seq[256*2048] | c[2048] | W_sum[8192*2048]
  float* seq  = (float*)d_ws;
  float* c    = seq + (size_t)SEQLEN * DD;
  float* Wsum = c + DD;
  const size_t need_wsum =
      ((size_t)SEQLEN * DD + DD + (size_t)4 * DD * DD) * sizeof(float);
  const bool have_wsum = ws_size >= need_wsum;

  if (have_wsum) {
    // 8192*2048 floats = 4,194,304 float4 -> 16384 blocks of 256 threads
    wsum_kernel<<<16384, 256, 0, stream>>>(
        (const float4*)W_ih, (const float4*)W_hh, (float4*)Wsum);
  }

  // Step 0: inp = x, h = 0 (so the W_hh term vanishes), c = 0 (first=1).
  lstm_step_kernel<<<256, 256, 0, stream>>>(
      W_ih, nullptr, x, b_ih, b_hh, c, seq, /*use_w2=*/0, /*first=*/1);

  // Steps 1..255: inp == h == previous output row of seq.
  for (int t = 1; t < SEQLEN; ++t) {
    const float* hprev = seq + (size_t)(t - 1) * DD;
    float* hout = seq + (size_t)t * DD;
    if (have_wsum) {
      lstm_step_kernel<<<256, 256, 0, stream>>>(
          Wsum, nullptr, hprev, b_ih, b_hh, c, hout, 0, 0);
    } else {
      lstm_step_kernel<<<256, 256, 0, stream>>>(
          W_ih, W_hh, hprev, b_ih, b_hh, c, hout, 1, 0);
    }
  }

  // Final projection GEMM: 1024 16x16 tiles / 8 waves per block -> 128 blocks.
  out_proj_kernel<<<128, 256, 0, stream>>>(seq, W_out, b_out, (float*)d_out);
}